// GTConvBank_45509473469008
// MI455X (gfx1250) — compile-verified
//
#include <hip/hip_runtime.h>
#include <hip/hip_bf16.h>

typedef __attribute__((ext_vector_type(2))) float v2f;
typedef __attribute__((ext_vector_type(8))) float v8f;
typedef __attribute__((ext_vector_type(4))) int   i4;
typedef __attribute__((ext_vector_type(4))) float f4;

// ---------------- Kernel 0: zero the output -----------------------------
__global__ void __launch_bounds__(256)
gt_zero(float* __restrict__ y, int n) {
    int i = blockIdx.x * blockDim.x + threadIdx.x;
    if (i < n) y[i] = 0.0f;
}

// ---------------- Kernel 1: Z = X @ h via V_WMMA_F32_16X16X4_F32 --------
// X: (N,16) row-major f32.  h: (16,K) row-major f32 (K<=16).  Z: (N,K) f32.
// One wave (32 lanes) computes a 16-row x 16-col tile; only cols < K stored.
// A 16x4 f32 layout: lanes 0-15 hold M=0..15 {K=c0,c0+1}, lanes 16-31 hold
// M=0..15 {K=c0+2,c0+3}  -> per-lane contiguous float2 load from X.
// B 4x16 f32 layout: V0 = rows {c0, c0+2} striped over lane halves, V1 =
// rows {c0+1, c0+3}.  Columns >= K are zero-masked (no EXEC divergence).
__global__ void __launch_bounds__(256)
gt_zgemm_wmma(const float* __restrict__ X, const float* __restrict__ h,
              float* __restrict__ Z, int N, int K, int ntiles) {
    int wave = blockIdx.x * (blockDim.x >> 5) + (threadIdx.x >> 5);
    if (wave >= ntiles) return;               // wave-uniform exit

    int lane = threadIdx.x & 31;
    int half = lane >> 4;                     // 0: lanes 0-15, 1: lanes 16-31
    int m    = lane & 15;

    int row = wave * 16 + m;
    if (row >= N) row = N - 1;                // clamp (stores are guarded)

    int   n     = m;                          // output column this lane owns
    int   nk    = (n < K) ? n : (K - 1);      // clamped, always in-bounds
    float nmask = (n < K) ? 1.0f : 0.0f;      // zero-pad columns >= K

    v8f acc = {};
    #pragma unroll
    for (int c0 = 0; c0 < 16; c0 += 4) {
        int c = c0 + 2 * half;
        v2f a;
        a.x = X[row * 16 + c];
        a.y = X[row * 16 + c + 1];
        v2f b;
        b.x = h[(c)     * K + nk] * nmask;
        b.y = h[(c + 1) * K + nk] * nmask;
        acc = __builtin_amdgcn_wmma_f32_16x16x4_f32(
                  /*neg_a=*/false, a, /*neg_b=*/false, b,
                  /*c_mod=*/(short)0, acc, /*reuse_a=*/false, /*reuse_b=*/false);
    }

    // D 16x16 f32: VGPR j holds row (j + 8*half), column = lane&15.
    if (n < K) {
        int rb = wave * 16 + 8 * half;
        #pragma unroll
        for (int j = 0; j < 8; ++j) {
            int r = rb + j;
            if (r < N) Z[r * K + n] = acc[j];
        }
    }
}

// ---------------- Kernel 2: edge gather * val -> atomic scatter ---------
// grid.y = k (0..K-1), grid.x covers E in chunks of 4 edges/thread.
// Edge arrays are streamed once (192 MB) -> non-temporal loads so the hot
// Z (2 MB) and y (400 KB) stay resident in the 192 MB L2.  Scatter uses
// unsafeAtomicAdd -> native global_atomic_add_f32 (no-return, L2-side),
// NOT a CAS loop.
__global__ void __launch_bounds__(256)
gt_edge_scatter(const int* __restrict__ rows, const int* __restrict__ cols,
                const float* __restrict__ vals, const float* __restrict__ Z,
                float* __restrict__ y, int E, int K) {
    int k = blockIdx.y;
    long long base = (long long)k * (long long)E;
    int i0 = (blockIdx.x * blockDim.x + threadIdx.x) * 4;

    if (i0 + 3 < E) {
        i4 r = __builtin_nontemporal_load((const i4*)(rows + base + i0));
        i4 c = __builtin_nontemporal_load((const i4*)(cols + base + i0));
        f4 v = __builtin_nontemporal_load((const f4*)(vals + base + i0));

        float z0 = Z[c.x * K + k];
        float z1 = Z[c.y * K + k];
        float z2 = Z[c.z * K + k];
        float z3 = Z[c.w * K + k];

        unsafeAtomicAdd(&y[r.x], v.x * z0);
        unsafeAtomicAdd(&y[r.y], v.y * z1);
        unsafeAtomicAdd(&y[r.z], v.z * z2);
        unsafeAtomicAdd(&y[r.w], v.w * z3);
    } else {
        for (int i = i0; i < E; ++i) {
            int   rr = rows[base + i];
            int   cc = cols[base + i];
            float vv = vals[base + i];
            unsafeAtomicAdd(&y[rr], vv * Z[cc * K + k]);
        }
    }
}

// ------------------------------------------------------------------------
extern "C" void kernel_launch(void* const* d_in, const int* in_sizes, int n_in,
                              void* d_out, int out_size, void* d_ws, size_t ws_size,
                              hipStream_t stream) {
    const float* X    = (const float*)d_in[0];   // (N, C)
    const int*   rows = (const int*)  d_in[1];   // (K, E)
    const int*   cols = (const int*)  d_in[2];   // (K, E)
    const float* vals = (const float*)d_in[3];   // (K, E)
    const float* h    = (const float*)d_in[4];   // (C, K)

    float* y = (float*)d_out;                    // (N,)
    float* Z = (float*)d_ws;                     // (N, K) scratch, N*K*4 bytes

    const int N = out_size;                      // 100000
    const int C = in_sizes[0] / N;               // 16
    const int K = in_sizes[4] / C;               // 5
    const int E = in_sizes[1] / K;               // 3200000
    (void)C; (void)n_in; (void)ws_size;

    // y = 0
    gt_zero<<<(N + 255) / 256, 256, 0, stream>>>(y, N);

    // Z = X @ h  (WMMA f32, one wave per 16-row tile, 8 waves/block)
    int ntiles = (N + 15) / 16;
    int nblk   = (ntiles + 7) / 8;
    gt_zgemm_wmma<<<nblk, 256, 0, stream>>>(X, h, Z, N, K, ntiles);

    // y[rows] += vals * Z[cols, k]
    int per_block = 256 * 4;
    dim3 grid((E + per_block - 1) / per_block, K);
    gt_edge_scatter<<<grid, 256, 0, stream>>>(rows, cols, vals, Z, y, E, K);
}